// Linear_73967926771855
// MI455X (gfx1250) — compile-verified
//
#include <hip/hip_runtime.h>
#include <hip/hip_bf16.h>

typedef float v2f __attribute__((ext_vector_type(2)));
typedef float v8f __attribute__((ext_vector_type(8)));

#define IN_F   128
#define OUT_F  128
#define LDS_S  132              // padded row stride (floats): 132 mod 64 = 4 -> conflict-free b64 reads
#define THREADS 256             // 8 wave32 waves
#define ROWS_PER_BLOCK 128      // 8 waves * 16 batch rows

// Y[b][n] = sum_k X[b][k] * W[n][k] + bias[n]
// V_WMMA_F32_16X16X4_F32: D(16x16,f32) = A(16x4) * B(4x16) + C
//
// A layout (32-bit A 16x4): lanes 0-15 row M=lane, VGPR0=K0/VGPR1=K1;
//                           lanes 16-31 row M=lane-16, VGPR0=K2/VGPR1=K3.
// B layout (mirror):        lanes 0-15 N=lane, VGPR0=K0/VGPR1=K1;
//                           lanes 16-31 N=lane-16, VGPR0=K2/VGPR1=K3.
// In row-major W[n][k] the per-lane B pair (K=4k+2*half, K=4k+2*half+1) is a
// contiguous float2 -> single ds_load_b64, no register shuffling.
__global__ __launch_bounds__(THREADS) void linear_wmma_f32(
    const float* __restrict__ X,
    const float* __restrict__ W,
    const float* __restrict__ bias,
    float* __restrict__ Y)
{
    extern __shared__ float wt[];   // wt[n*LDS_S + k] = W[n][k], OUT_F x LDS_S

    const int tid = threadIdx.x;

    // Stage W into LDS: coalesced on both sides (i = n*128 + k).
    for (int i = tid; i < IN_F * OUT_F; i += THREADS) {
        const int n = i >> 7;
        const int k = i & 127;
        wt[n * LDS_S + k] = W[i];
    }
    __syncthreads();

    const int wave = tid >> 5;           // 0..7
    const int lane = tid & 31;
    const int ln   = lane & 15;          // N (for B/D) / M (for A) sub-index
    const int half = lane >> 4;          // 0: K+0/K+1, 1: K+2/K+3

    const long row0 = (long)blockIdx.x * ROWS_PER_BLOCK + (long)wave * 16;

    // A source: row = row0 + ln, starting K offset = 2*half.
    const float* xrow = X + (row0 + ln) * (long)IN_F + 2 * half;
    // B source: wt[(n0+ln)][4k + 2*half], single base + immediate ds offsets.
    const float* wrow = wt + ln * LDS_S + 2 * half;

    // 8 accumulators (one per 16-col output tile), pre-loaded with bias.
    v8f c[8];
#pragma unroll
    for (int nt = 0; nt < 8; ++nt) {
        const float bval = bias[nt * 16 + ln];
#pragma unroll
        for (int i = 0; i < 8; ++i) c[nt][i] = bval;
    }

#pragma unroll
    for (int k = 0; k < 32; ++k) {
        const v2f av = *(const v2f*)(xrow + 4 * k);          // global_load_b64
#pragma unroll
        for (int nt = 0; nt < 8; ++nt) {
            const v2f bv = *(const v2f*)(wrow + nt * 16 * LDS_S + 4 * k);  // ds_load_b64
            c[nt] = __builtin_amdgcn_wmma_f32_16x16x4_f32(
                        /*neg_a=*/false, av,
                        /*neg_b=*/false, bv,
                        /*c_mod=*/(short)0, c[nt],
                        /*reuse_a=*/false, /*reuse_b=*/false);
        }
    }

    // D layout: VGPR v -> row v (lanes 0-15) / row v+8 (lanes 16-31), col = n0+ln.
    float* out = Y + row0 * (long)OUT_F;
    const int rofs = half * 8;
#pragma unroll
    for (int nt = 0; nt < 8; ++nt) {
        const int n0 = nt * 16;
#pragma unroll
        for (int v = 0; v < 8; ++v) {
            out[(long)(v + rofs) * OUT_F + n0 + ln] = c[nt][v];
        }
    }
}

extern "C" void kernel_launch(void* const* d_in, const int* in_sizes, int n_in,
                              void* d_out, int out_size, void* d_ws, size_t ws_size,
                              hipStream_t stream) {
    const float* X    = (const float*)d_in[0];   // enc_x  [BATCH, 128]
    const float* W    = (const float*)d_in[1];   // weight [128, 128]
    const float* bias = (const float*)d_in[2];   // bias   [128]
    float* Y          = (float*)d_out;           // [BATCH, 128]

    const int batch  = in_sizes[0] / IN_F;
    const int blocks = batch / ROWS_PER_BLOCK;   // 4096

    const size_t lds_bytes = (size_t)OUT_F * LDS_S * sizeof(float);  // 67,584 B
    linear_wmma_f32<<<blocks, THREADS, lds_bytes, stream>>>(X, W, bias, Y);
}